// YOLOv8Loss_51616916963406
// MI455X (gfx1250) — compile-verified
//
#include <hip/hip_runtime.h>
#include <hip/hip_bf16.h>

typedef __attribute__((ext_vector_type(2))) float v2f;
typedef __attribute__((ext_vector_type(8))) float v8f;

// ---------------------------------------------------------------------------
// Wave-level dual reduction via V_WMMA_F32_16X16X4_F32.
//
// A (16x4 f32, 2 VGPRs/lane):
//   lanes 0-15 : VGPR0 -> A[m][0] = pbox[m],  VGPR1 -> A[m][1] = pcls[m]
//   lanes 16-31: VGPR0 -> A[m][2] = pbox[16+m], VGPR1 -> A[m][3] = pcls[16+m]
// B (4x16, 2 VGPRs/lane): selector so that
//   B[0][n]=B[1][n] = (n<8),  B[2][n]=B[3][n] = (n>=8)
// => D[m][n<8]  = pbox[m]+pcls[m]
//    D[m][n>=8] = pbox[16+m]+pcls[16+m]
// In-lane sum of the 8 D VGPRs gives half-sums; shfl_xor(16) completes them.
// Result: lanes with (lane&15)<8 hold pbox+pcls summed over lanes 0-15, others over lanes 16-31.
// ---------------------------------------------------------------------------
__device__ inline void wmma_wave_reduce2(float pbox, float pcls,
                                         float& sum_box, float& sum_cls) {
  const int lane = threadIdx.x & 31;
  v2f a;
  a.x = pbox;
  a.y = pcls;

  const int n = lane & 15;
  const float bval = ((n < 8) == (lane < 16)) ? 1.0f : 0.0f;
  v2f b;
  b.x = bval;
  b.y = bval;

  v8f c = {};
  // 8 args: (neg_a, A, neg_b, B, c_mod, C, reuse_a, reuse_b)
  v8f d = __builtin_amdgcn_wmma_f32_16x16x4_f32(
      /*neg_a=*/false, a, /*neg_b=*/false, b,
      /*c_mod=*/(short)0, c, /*reuse_a=*/false, /*reuse_b=*/false);

  float s = d[0] + d[1] + d[2] + d[3] + d[4] + d[5] + d[6] + d[7];
  s += __shfl_xor(s, 16, 32);
  sum_box = s;  // valid where (lane & 15) < 8  (e.g. lane 0)
  sum_cls = s;  // valid where (lane & 15) >= 8 (e.g. lane 8)
}

// ---------------------------------------------------------------------------
// Main kernel: one wave per (level, batch, target) gather point. 1536 waves.
// ---------------------------------------------------------------------------
__global__ __launch_bounds__(256) void yolo_main_kernel(
    const float* __restrict__ p0, const float* __restrict__ p1,
    const float* __restrict__ p2, const float* __restrict__ targets,
    float* __restrict__ ws) {
  const int wave = threadIdx.x >> 5;
  const int lane = threadIdx.x & 31;
  const int point = blockIdx.x * 8 + wave;  // 0..1535
  const int level = point / 512;
  const int rem = point - level * 512;
  const int b = rem >> 5;
  const int t_idx = rem & 31;

  const float* pred;
  int H, W;
  float stride;
  if (level == 0) {
    pred = p0; H = 80; W = 80; stride = 8.0f;
  } else if (level == 1) {
    pred = p1; H = 40; W = 40; stride = 16.0f;
  } else {
    pred = p2; H = 20; W = 20; stride = 32.0f;
  }

  const float* t = targets + (size_t)(b * 32 + t_idx) * 5;
  const float x1 = t[0], y1 = t[1], x2 = t[2], y2 = t[3];
  const int cls = (int)t[4];
  const float cx = (x1 + x2) * 0.5f;
  const float cy = (y1 + y2) * 0.5f;
  const int gi = (int)fminf(fmaxf(cy / stride, 0.0f), (float)(H - 1));
  const int gj = (int)fminf(fmaxf(cx / stride, 0.0f), (float)(W - 1));

  const long HW = (long)H * W;
  const float* base = pred + (long)b * 144 * HW + (long)gi * W + gj;

  float pbox = 0.0f, pcls = 0.0f;
  for (int ch = lane; ch < 144; ch += 32) {
    const float x = base[(long)ch * HW];
    if (ch < 64) {
      pbox += x;
    } else {
      const float label = ((ch - 64) == cls) ? 1.0f : 0.0f;
      const float bce = fmaxf(x, 0.0f) - x * label + log1pf(expf(-fabsf(x)));
      pcls += bce;
    }
  }

  float sum_box, sum_cls;
  wmma_wave_reduce2(pbox, pcls, sum_box, sum_cls);

  if (lane == 0) ws[point] = sum_box;         // 32 box feats + 48 BCE terms (lanes 0-15)
  if (lane == 8) ws[1536 + point] = sum_cls;  // 32 box feats + 32 BCE terms (lanes 16-31)
}

// ---------------------------------------------------------------------------
// Finalize: deterministic reduction of 1536 per-point partials -> 4 outputs.
// ---------------------------------------------------------------------------
__global__ __launch_bounds__(256) void yolo_finalize_kernel(
    const float* __restrict__ ws, float* __restrict__ out) {
  __shared__ float sb[256];
  __shared__ float sc[256];
  float ab = 0.0f, ac = 0.0f;
  for (int i = threadIdx.x; i < 1536; i += 256) {
    ab += ws[i];
    ac += ws[1536 + i];
  }
  sb[threadIdx.x] = ab;
  sc[threadIdx.x] = ac;
  __syncthreads();
  for (int s = 128; s > 0; s >>= 1) {
    if ((int)threadIdx.x < s) {
      sb[threadIdx.x] += sb[threadIdx.x + s];
      sc[threadIdx.x] += sc[threadIdx.x + s];
    }
    __syncthreads();
  }
  if (threadIdx.x == 0) {
    const float inv_num = 1.0f / 512.0f;  // B*N = 16*32
    const float loss_box = (-0.1f * sb[0] / 64.0f) * inv_num;
    const float loss_cls = (sc[0] / 80.0f) * inv_num;
    const float loss_dfl = 0.0f;
    const float total = 7.5f * loss_box + 0.5f * loss_cls + 1.5f * loss_dfl;
    out[0] = total;
    out[1] = loss_box;
    out[2] = loss_cls;
    out[3] = loss_dfl;
  }
}

extern "C" void kernel_launch(void* const* d_in, const int* in_sizes, int n_in,
                              void* d_out, int out_size, void* d_ws,
                              size_t ws_size, hipStream_t stream) {
  const float* p0 = (const float*)d_in[0];
  const float* p1 = (const float*)d_in[1];
  const float* p2 = (const float*)d_in[2];
  const float* targets = (const float*)d_in[3];
  float* out = (float*)d_out;
  float* ws = (float*)d_ws;  // 2 * 1536 floats, fully overwritten each call

  // 1536 gather points, 8 waves (256 threads) per block -> 192 blocks.
  yolo_main_kernel<<<192, 256, 0, stream>>>(p0, p1, p2, targets, ws);
  yolo_finalize_kernel<<<1, 256, 0, stream>>>(ws, out);
}